// MultiBoxLoss_51917564674061
// MI455X (gfx1250) — compile-verified
//
#include <hip/hip_runtime.h>
#include <hip/hip_bf16.h>

#define THREADS 256
#define MAXP    8732
#define MM      16
#define NB2     512

typedef __attribute__((ext_vector_type(2))) float v2f;
typedef __attribute__((ext_vector_type(8))) float v8f;

// Exact f32 wave(32) sum using the matrix pipe: D = A(16x4) * ones(4x16).
// Each lane contributes its partial in A K-slot 0 (a[0]); a[1]=0. With B all
// ones, every D column equals the 16 row-sums, and summing the 8 C/D VGPRs at
// lane j plus lane j^16 covers all 16 rows => total of all 32 lane values.
__device__ __forceinline__ float wave_sum32(float v) {
#if __has_builtin(__builtin_amdgcn_wmma_f32_16x16x4_f32)
  v2f a; a[0] = v;    a[1] = 0.0f;
  v2f b; b[0] = 1.0f; b[1] = 1.0f;
  v8f c = {};
  c = __builtin_amdgcn_wmma_f32_16x16x4_f32(false, a, false, b, (short)0, c, false, false);
  float s = c[0] + c[1] + c[2] + c[3] + c[4] + c[5] + c[6] + c[7];
  s += __shfl_xor(s, 16, 32);
  return s;
#else
  for (int off = 16; off > 0; off >>= 1) v += __shfl_xor(v, off, 32);
  return v;
#endif
}

// ---------------------------------------------------------------------------
// Kernel 1: per-image prior<->box matching. One block per batch image.
// Produces packed (label<<8 | object) per prior and n_pos per image.
// ---------------------------------------------------------------------------
__global__ __launch_bounds__(THREADS) void match_kernel(
    const float4* __restrict__ boxes,    // [B, 16] xy
    const int*    __restrict__ labels,   // [B, 16]
    const float4* __restrict__ priors,   // [P] cxcy
    int*          __restrict__ match,    // [B, P] packed
    int*          __restrict__ n_pos,    // [B]
    int P)
{
  __shared__ float         s_iou[MAXP];
  __shared__ unsigned char s_obj[MAXP];
  __shared__ float4 sb[MM];
  __shared__ float  sa[MM];
  __shared__ int    slab[MM];
  __shared__ float  riou[THREADS];
  __shared__ int    rp[THREADS];
  __shared__ int    bestp[MM];

  const int b = blockIdx.x;
  const int tid = threadIdx.x;

  if (tid < MM) {
    float4 bx = boxes[b * MM + tid];
    sb[tid] = bx;
    sa[tid] = (bx.z - bx.x) * (bx.w - bx.y);
    slab[tid] = labels[b * MM + tid];
  }
  __syncthreads();

  float bi[MM]; int bp[MM];
#pragma unroll
  for (int m = 0; m < MM; ++m) { bi[m] = -1.0f; bp[m] = 0; }

  for (int p = tid; p < P; p += THREADS) {
    float4 pr = priors[p];
    float hx = pr.z * 0.5f, hy = pr.w * 0.5f;
    float px1 = pr.x - hx, py1 = pr.y - hy;
    float px2 = pr.x + hx, py2 = pr.y + hy;
    float pa = (px2 - px1) * (py2 - py1);
    float best = -1.0f; int bo = 0;
#pragma unroll
    for (int m = 0; m < MM; ++m) {
      float4 bx = sb[m];
      float iw = fmaxf(fminf(bx.z, px2) - fmaxf(bx.x, px1), 0.0f);
      float ih = fmaxf(fminf(bx.w, py2) - fmaxf(bx.y, py1), 0.0f);
      float inter = iw * ih;
      float iou = inter / (sa[m] + pa - inter);
      if (iou > best)  { best = iou; bo = m; }      // first max over m (like argmax)
      if (iou > bi[m]) { bi[m] = iou; bp[m] = p; }  // earliest p kept within thread
    }
    s_iou[p] = best;
    s_obj[p] = (unsigned char)bo;
  }
  __syncthreads();

  // per-object argmax over P (tie-break: smallest p, matching jnp.argmax)
  for (int m = 0; m < MM; ++m) {
    riou[tid] = bi[m]; rp[tid] = bp[m];
    __syncthreads();
    for (int s = THREADS / 2; s > 0; s >>= 1) {
      if (tid < s) {
        float oi = riou[tid + s]; int op = rp[tid + s];
        if (oi > riou[tid] || (oi == riou[tid] && op < rp[tid])) { riou[tid] = oi; rp[tid] = op; }
      }
      __syncthreads();
    }
    if (tid == 0) bestp[m] = rp[0];
    __syncthreads();
  }

  // force-assign (serial so duplicate best-priors resolve "last wins" deterministically)
  if (tid == 0) {
    for (int m = 0; m < MM; ++m) {
      int q = bestp[m];
      s_obj[q] = (unsigned char)m;
      s_iou[q] = 1.0f;
    }
  }
  __syncthreads();

  int cnt = 0;
  for (int p = tid; p < P; p += THREADS) {
    int o = s_obj[p];
    int lab = (s_iou[p] < 0.5f) ? 0 : slab[o];
    match[(size_t)b * P + p] = (lab << 8) | o;
    cnt += (lab != 0) ? 1 : 0;
  }
  rp[tid] = cnt; __syncthreads();
  for (int s = THREADS / 2; s > 0; s >>= 1) { if (tid < s) rp[tid] += rp[tid + s]; __syncthreads(); }
  if (tid == 0) n_pos[b] = rp[0];
}

// ---------------------------------------------------------------------------
// Kernel 2: streaming pass over scores (94 MB) + locs. HBM-bound.
// Computes conf = logsumexp(s) - s[label]; accumulates positive conf and
// positive L1 loc loss (deterministic per-block partials); writes conf_neg.
// Wave reductions use V_WMMA_F32_16X16X4_F32.
// ---------------------------------------------------------------------------
__global__ __launch_bounds__(THREADS) void conf_kernel(
    const float*  __restrict__ locs,      // [B,P,4]
    const float*  __restrict__ scores,    // [B,P,C]
    const float4* __restrict__ boxes,     // [B,16] xy
    const float4* __restrict__ priors,    // [P] cxcy
    const int*    __restrict__ match,     // [B,P]
    float*        __restrict__ conf_neg,  // [B,P]
    float*        __restrict__ partial,   // [NB2*2]
    int B, int P, int C)
{
  __shared__ float wconf[THREADS / 32];
  __shared__ float wloc[THREADS / 32];
  const int tid = threadIdx.x;
  const long long total  = (long long)B * P;
  const long long stride = (long long)gridDim.x * blockDim.x;

  float conf_pos = 0.0f, loc_abs = 0.0f;

  for (long long idx = (long long)blockIdx.x * blockDim.x + tid; idx < total; idx += stride) {
    int mt  = match[idx];
    int obj = mt & 0xFF;
    int cls = mt >> 8;

    const float* sp = scores + idx * C;
    __builtin_prefetch(sp + stride * C, 0, 0);  // -> global_prefetch_b8
    float mx = sp[0];
    for (int c = 1; c < C; ++c) mx = fmaxf(mx, sp[c]);
    float se = 0.0f;
    for (int c = 0; c < C; ++c) se += __expf(sp[c] - mx);
    float conf = mx + __logf(se) - sp[cls];

    if (cls != 0) {
      conf_pos += conf;
      conf_neg[idx] = 0.0f;
      int b = (int)(idx / P);
      int p = (int)(idx - (long long)b * P);
      float4 bx = boxes[b * MM + obj];
      float4 pr = priors[p];
      float cx = (bx.x + bx.z) * 0.5f, cy = (bx.y + bx.w) * 0.5f;
      float w  = bx.z - bx.x,          h  = bx.w - bx.y;
      float gx = (cx - pr.x) / (pr.z * 0.1f);
      float gy = (cy - pr.y) / (pr.w * 0.1f);
      float gw = __logf(w / pr.z) * 5.0f;
      float gh = __logf(h / pr.w) * 5.0f;
      const float* pl = locs + idx * 4;
      loc_abs += fabsf(pl[0] - gx) + fabsf(pl[1] - gy) +
                 fabsf(pl[2] - gw) + fabsf(pl[3] - gh);
    } else {
      // conf >= 0 mathematically; clamp fp slop so uint radix-select ordering holds
      conf_neg[idx] = fmaxf(conf, 0.0f);
    }
  }

  float cs = wave_sum32(conf_pos);  // v_wmma_f32_16x16x4_f32
  float ls = wave_sum32(loc_abs);   // v_wmma_f32_16x16x4_f32
  if ((tid & 31) == 0) { wconf[tid >> 5] = cs; wloc[tid >> 5] = ls; }
  __syncthreads();
  if (tid == 0) {
    float a = 0.0f, l = 0.0f;
    for (int w = 0; w < THREADS / 32; ++w) { a += wconf[w]; l += wloc[w]; }
    partial[2 * blockIdx.x]     = a;
    partial[2 * blockIdx.x + 1] = l;
  }
}

// ---------------------------------------------------------------------------
// Kernel 3: hard-negative mining. One block per image; row staged in LDS
// (35 KB of 320 KB/WGP). Exact top-K sum via 32-pass binary radix select of
// the K-th largest value (all values >= 0 so float bits order as uints):
//   sum = sum(v > t) + t * (K - count(v > t))   (== descending-sort top-K sum)
// ---------------------------------------------------------------------------
__global__ __launch_bounds__(THREADS) void mine_kernel(
    const float* __restrict__ conf_neg,  // [B,P]
    const int*   __restrict__ n_pos,     // [B]
    float*       __restrict__ hardneg,   // [B]
    int P)
{
  __shared__ float s_v[MAXP];
  __shared__ int   redi[THREADS];
  __shared__ float redf[THREADS];
  const int b = blockIdx.x;
  const int tid = threadIdx.x;

  for (int i = tid; i < P; i += THREADS) s_v[i] = conf_neg[(size_t)b * P + i];
  __syncthreads();

  int K0 = n_pos[b] * 3;
  if (K0 > P) K0 = P;
  if (K0 <= 0) { if (tid == 0) hardneg[b] = 0.0f; return; }

  unsigned int prefix = 0u;
  int K = K0;
  for (int bit = 31; bit >= 0; --bit) {
    unsigned int cand = prefix | (1u << bit);
    int cnt = 0;
    for (int i = tid; i < P; i += THREADS) {
      unsigned int u = __float_as_uint(s_v[i]);
      cnt += ((u >> bit) == (cand >> bit)) ? 1 : 0;
    }
    redi[tid] = cnt; __syncthreads();
    for (int s = THREADS / 2; s > 0; s >>= 1) { if (tid < s) redi[tid] += redi[tid + s]; __syncthreads(); }
    int c = redi[0];
    __syncthreads();
    if (c >= K) prefix = cand; else K -= c;   // uniform across block
  }

  float t = __uint_as_float(prefix);  // exact K-th largest
  float sgt = 0.0f; int cgt = 0;
  for (int i = tid; i < P; i += THREADS) {
    float v = s_v[i];
    if (v > t) { sgt += v; cgt++; }
  }
  redf[tid] = sgt; redi[tid] = cgt; __syncthreads();
  for (int s = THREADS / 2; s > 0; s >>= 1) {
    if (tid < s) { redf[tid] += redf[tid + s]; redi[tid] += redi[tid + s]; }
    __syncthreads();
  }
  if (tid == 0) hardneg[b] = redf[0] + t * (float)(K0 - redi[0]);
}

// ---------------------------------------------------------------------------
// Kernel 4: deterministic final combine -> scalar loss.
// ---------------------------------------------------------------------------
__global__ void finalize_kernel(
    const float* __restrict__ partial, const float* __restrict__ hardneg,
    const int* __restrict__ n_pos, float* __restrict__ out, int nb2, int B)
{
  if (threadIdx.x == 0 && blockIdx.x == 0) {
    float cps = 0.0f, labs = 0.0f;
    for (int i = 0; i < nb2; ++i) { cps += partial[2 * i]; labs += partial[2 * i + 1]; }
    float hn = 0.0f; int npt = 0;
    for (int b = 0; b < B; ++b) { hn += hardneg[b]; npt += n_pos[b]; }
    float conf_loss = (hn + cps) / (float)npt;
    int d = npt * 4; if (d < 1) d = 1;
    out[0] = conf_loss + labs / (float)d;
  }
}

extern "C" void kernel_launch(void* const* d_in, const int* in_sizes, int n_in,
                              void* d_out, int out_size, void* d_ws, size_t ws_size,
                              hipStream_t stream) {
  (void)n_in; (void)out_size;
  const float* predicted_locs   = (const float*)d_in[0];
  const float* predicted_scores = (const float*)d_in[1];
  const float* boxes            = (const float*)d_in[2];
  const int*   labels           = (const int*)d_in[3];
  const float* priors           = (const float*)d_in[4];

  const int B = in_sizes[3] / MM;                       // labels: [B,16]
  const int P = in_sizes[4] / 4;                        // priors: [P,4]
  const int C = (int)((long long)in_sizes[1] / ((long long)B * P));
  if (P > MAXP || B <= 0 || C <= 0) return;

  size_t need = (size_t)B * P * sizeof(int)      // match
              + (size_t)B * P * sizeof(float)    // conf_neg
              + (size_t)B * sizeof(int)          // n_pos
              + (size_t)B * sizeof(float)        // hardneg
              + (size_t)NB2 * 2 * sizeof(float); // partials
  if (ws_size < need) return;

  char* ws = (char*)d_ws;
  int*   match_arr = (int*)ws;    ws += (size_t)B * P * sizeof(int);
  float* conf_neg  = (float*)ws;  ws += (size_t)B * P * sizeof(float);
  int*   n_pos     = (int*)ws;    ws += (size_t)B * sizeof(int);
  float* hardneg   = (float*)ws;  ws += (size_t)B * sizeof(float);
  float* partial   = (float*)ws;

  match_kernel<<<B, THREADS, 0, stream>>>(
      (const float4*)boxes, labels, (const float4*)priors, match_arr, n_pos, P);

  conf_kernel<<<NB2, THREADS, 0, stream>>>(
      predicted_locs, predicted_scores, (const float4*)boxes,
      (const float4*)priors, match_arr, conf_neg, partial, B, P, C);

  mine_kernel<<<B, THREADS, 0, stream>>>(conf_neg, n_pos, hardneg, P);

  finalize_kernel<<<1, 32, 0, stream>>>(partial, hardneg, n_pos, (float*)d_out, NB2, B);
}